// Paired_Windows_Attention_49452253446325
// MI455X (gfx1250) — compile-verified
//
#include <hip/hip_runtime.h>
#include <hip/hip_bf16.h>
#include <math.h>

typedef __attribute__((ext_vector_type(2))) float v2f;
typedef __attribute__((ext_vector_type(8))) float v8f;

#define NWIN_TOT   4680                 // 16^3 + 8^3 + 4^3 + 2^3
#define WIN_ELEMS  216                  // 27 * 8
#define TOT_PER_TENSOR (4*2*4680*216)   // 8,087,040 floats per gathered tensor

__device__ __forceinline__ v8f wmma_f32_16x16x4(v2f a, v2f b, v8f c) {
  // D = A(16x4,f32) * B(4x16,f32) + C(16x16,f32)
  return __builtin_amdgcn_wmma_f32_16x16x4_f32(false, a, false, b, (short)0, c, false, false);
}

// ---------------------------------------------------------------------------
// Phase A: branch-wise max-pool gather.
// One thread per pooled output element; thread order == output flat index
// (b, H, N, L=27, C=8) -> perfectly coalesced stores, contiguous pooled reads.
// ---------------------------------------------------------------------------
__global__ __launch_bounds__(256) void gather_pool_kernel(
    const float* __restrict__ q, const float* __restrict__ k, const float* __restrict__ v,
    float* __restrict__ qg, float* __restrict__ kg, float* __restrict__ vg) {
  long long idx = (long long)blockIdx.x * blockDim.x + threadIdx.x;
  const long long TOT = (long long)TOT_PER_TENSOR;
  if (idx >= 3 * TOT) return;
  int t   = (int)(idx / TOT);
  int rem = (int)(idx - (long long)t * TOT);

  int c   = rem & 7;
  int tmp = rem >> 3;
  int l   = tmp % 27;  tmp /= 27;
  int n   = tmp % NWIN_TOT; tmp /= NWIN_TOT;
  int head = tmp & 1;
  int b    = tmp >> 1;

  int br, Nh, n0;
  if      (n < 4096) { br = 0; Nh = 16; n0 = n;        }
  else if (n < 4608) { br = 1; Nh = 8;  n0 = n - 4096; }
  else if (n < 4672) { br = 2; Nh = 4;  n0 = n - 4608; }
  else               { br = 3; Nh = 2;  n0 = n - 4672; }

  int Wd = n0 % Nh; int t2 = n0 / Nh;
  int Ww = t2 % Nh; int Wh = t2 / Nh;
  int wh = l / 9, ww = (l / 3) % 3, wd = l % 3;

  int sw = 1 << br;        // SMALL = [1,2,4,8]
  int bw = 3 * sw;         // BIG   = [3,6,12,24]
  int h0 = Wh * bw + wh * sw;
  int w0 = Ww * bw + ww * sw;
  int d0 = Wd * bw + wd * sw;
  int chIn = br * 16 + head * 8 + c;   // (G, BSWIN, H, C) channel order

  const float* src = (t == 0) ? q : ((t == 1) ? k : v);
  src += ((size_t)(b * 64 + chIn)) * 110592;   // 48^3

  float mx = -INFINITY;
  for (int ph = 0; ph < sw; ++ph)
    for (int pw = 0; pw < sw; ++pw) {
      int base = (h0 + ph) * 2304 + (w0 + pw) * 48 + d0;
      for (int pd = 0; pd < sw; ++pd)
        mx = fmaxf(mx, src[base + pd]);
    }

  float* dst = (t == 0) ? qg : ((t == 1) ? kg : vg);
  dst[rem] = mx;
}

// ---------------------------------------------------------------------------
// Phase B: one wave32 per window; all heavy math on the WMMA pipe.
//   scores S = q*k^T            : 2x2 tiles x 2 k-steps          =  8 WMMA
//   E = exp2(S*log2e/sqrt(8)) -> LDS 32x28 tile (branch-free stores;
//     rows>=27 and col 27 hold garbage that only ever multiplies zeros)
//   Ev = E*v, rowsum = E*ones, Bv = bias*v : 3 chains x 2 Mtiles x 7 = 42 WMMA
//   out = Ev * rcp(rowsum) + Bv   (rowsum broadcast by WMMA into the exact
//                                  C-layout slot of Ev -> elementwise combine)
// Softmax max-subtraction dropped: |S/sqrt(8)| <~ 15, exp cannot overflow fp32,
// result is mathematically identical. No shuffles, no exec-mask branches.
// ---------------------------------------------------------------------------
__global__ __launch_bounds__(128) void win_attn_wmma_kernel(
    const float* __restrict__ qg, const float* __restrict__ kg, const float* __restrict__ vg,
    const float* __restrict__ rpb, float* __restrict__ attn) {
  __shared__ float wbuf[4][32 * 28];    // per-wave E tile, stride 28
  __shared__ float biasF[2][27 * 28];   // rel-bias (H,27,27), stride 28, col 27 = 0

  int tid = threadIdx.x;
  // Expand rpb_table(125,2) -> biasF once per block.
  for (int i = tid; i < 2 * 27 * 28; i += 128) {
    int h   = i / (27 * 28);
    int rcd = i - h * (27 * 28);
    int mm  = rcd / 28, nn = rcd % 28;
    float bv = 0.f;
    if (nn < 27) {
      int dh = mm / 9       - nn / 9       + 2;
      int dw = (mm / 3) % 3 - (nn / 3) % 3 + 2;
      int dd = mm % 3       - nn % 3       + 2;
      bv = rpb[((dh * 5 + dw) * 5 + dd) * 2 + h];
    }
    (&biasF[0][0])[i] = bv;
  }
  __syncthreads();

  int wave = tid >> 5, lane = tid & 31;
  int win  = blockIdx.x * 4 + wave;     // (b*2+head)*4680 + n
  int head = (win / NWIN_TOT) & 1;

  const float* qw = qg + (size_t)win * WIN_ELEMS;
  const float* kw = kg + (size_t)win * WIN_ELEMS;
  const float* vw = vg + (size_t)win * WIN_ELEMS;
  float*       ow = attn + (size_t)win * WIN_ELEMS;

  int r  = lane & 15;   // row/col within 16-tile
  int hh = lane >> 4;   // K-half selector for A/B fragments
  float* wb = wbuf[wave];

  // ---- scores S = q * k^T  (A-frag of q, B-frag of k share the same layout) ----
  v2f aq[2][2], bk[2][2];
#pragma unroll
  for (int tile = 0; tile < 2; ++tile) {
    int row  = tile * 16 + r;
    bool ok  = row < 27;
    int rowc = ok ? row : 26;           // clamped address, unconditional load
#pragma unroll
    for (int ks = 0; ks < 2; ++ks) {
      int c0 = ks * 4 + hh * 2;         // even -> 8B aligned
      v2f a  = *(const v2f*)(qw + rowc * 8 + c0);
      v2f bb = *(const v2f*)(kw + rowc * 8 + c0);
      aq[tile][ks].x = ok ? a.x  : 0.f;
      aq[tile][ks].y = ok ? a.y  : 0.f;
      bk[tile][ks].x = ok ? bb.x : 0.f;
      bk[tile][ks].y = ok ? bb.y : 0.f;
    }
  }
  v8f s[2][2];
#pragma unroll
  for (int mt = 0; mt < 2; ++mt)
#pragma unroll
    for (int nt = 0; nt < 2; ++nt) {
      v8f acc = {};
      acc = wmma_f32_16x16x4(aq[mt][0], bk[nt][0], acc);
      acc = wmma_f32_16x16x4(aq[mt][1], bk[nt][1], acc);
      s[mt][nt] = acc;
    }

  // ---- E = exp2(S * log2(e)/sqrt(8)) into LDS tile, fully branch-free ----
  const float rs2 = 0.35355339059327373f * 1.4426950408889634f;
  bool nv1 = (r < 11);                  // col 16+r < 27
  int col1 = nv1 ? 16 + r : 27;         // invalid lanes spill into pad col 27
#pragma unroll
  for (int mt = 0; mt < 2; ++mt) {
#pragma unroll
    for (int e = 0; e < 8; ++e) {
      int m = mt * 16 + e + 8 * hh;     // <= 31; rows >= 27 are don't-care
      wb[m * 28 + r]    = __builtin_amdgcn_exp2f(s[mt][0][e] * rs2);
      wb[m * 28 + col1] = __builtin_amdgcn_exp2f(s[mt][1][e] * rs2);
    }
  }

  // ---- v B-fragments (27x8, K padded to 28; col clamped, pad zeroed) ----
  bool ncol = (r < 8);
  int  nc   = ncol ? r : 7;
  v2f bvf[7];
#pragma unroll
  for (int ks = 0; ks < 7; ++ks) {
    int k0 = ks * 4 + hh * 2;            // <= 26, always a valid row
    int k1 = k0 + 1;                     // == 27 only when ks==6 && hh==1
    int k1c = (k1 > 26) ? 26 : k1;
    float x = vw[k0  * 8 + nc];
    float y = vw[k1c * 8 + nc];
    bvf[ks].x = ncol              ? x : 0.f;
    bvf[ks].y = (ncol && k1 < 27) ? y : 0.f;
  }

  // ---- Ev = E*v, rowsum = E*ones, Bv = bias*v; combine elementwise ----
  const float* bh = &biasF[head][0];
#pragma unroll
  for (int mt = 0; mt < 2; ++mt) {
    v8f accEv = {}, accSum = {}, accBv = {};
    int arow = mt * 16 + r;                     // <= 31 (pad rows: don't-care)
    int brow = (arow > 26) ? 26 : arow;         // biasF has 27 rows
#pragma unroll
    for (int ks = 0; ks < 7; ++ks) {
      int k0 = ks * 4 + hh * 2;
      v2f aE, aB, on;
      aE.x = wb[arow * 28 + k0];
      aE.y = wb[arow * 28 + k0 + 1];
      aB.x = bh[brow * 28 + k0];
      aB.y = bh[brow * 28 + k0 + 1];
      on.x = 1.f;
      on.y = (k0 + 1 < 27) ? 1.f : 0.f;         // mask the k==27 pad slot
      accEv  = wmma_f32_16x16x4(aE, bvf[ks], accEv);
      accSum = wmma_f32_16x16x4(aE, on,      accSum);
      accBv  = wmma_f32_16x16x4(aB, bvf[ks], accBv);
    }
#pragma unroll
    for (int e = 0; e < 8; ++e) {
      int row = mt * 16 + e + 8 * hh;
      if (row < 27 && ncol)
        ow[row * 8 + r] = accEv[e] * __builtin_amdgcn_rcpf(accSum[e]) + accBv[e];
    }
  }
}

// ---------------------------------------------------------------------------
// Phase C: trilinear upsample scatter back to (4, 64, 48, 48, 48).
// One thread per output voxel, z innermost -> coalesced stores.
// ---------------------------------------------------------------------------
__global__ __launch_bounds__(256) void scatter_upsample_kernel(
    const float* __restrict__ attn, float* __restrict__ out) {
  long long idx = (long long)blockIdx.x * blockDim.x + threadIdx.x;
  const long long TOTO = (long long)4 * 64 * 110592;
  if (idx >= TOTO) return;

  int z = (int)(idx % 48);
  long long t = idx / 48;
  int y  = (int)(t % 48); t /= 48;
  int x  = (int)(t % 48); t /= 48;
  int ch = (int)(t % 64);
  int b  = (int)(t / 64);

  int br = ch >> 4, head = (ch >> 3) & 1, c = ch & 7;
  int sw = 1 << br, S = 3 * sw, Nh = 48 / S;

  int Wh = x / S, jh = x % S;
  int Ww = y / S, jw = y % S;
  int Wd = z / S, jd = z % S;
  int noff = (br == 0) ? 0 : ((br == 1) ? 4096 : ((br == 2) ? 4608 : 4672));
  int n = noff + (Wh * Nh + Ww) * Nh + Wd;

  const float* aw = attn + ((size_t)(b * 2 + head) * NWIN_TOT + n) * WIN_ELEMS + c;

  float step = 2.0f / (float)(S - 1);   // linspace(0, 2, S)
  float ph = jh * step; int i0h = (int)ph; if (i0h > 2) i0h = 2;
  int i1h = (i0h + 1 > 2) ? 2 : i0h + 1; float fh = ph - (float)i0h;
  float pw = jw * step; int i0w = (int)pw; if (i0w > 2) i0w = 2;
  int i1w = (i0w + 1 > 2) ? 2 : i0w + 1; float fw = pw - (float)i0w;
  float pd = jd * step; int i0d = (int)pd; if (i0d > 2) i0d = 2;
  int i1d = (i0d + 1 > 2) ? 2 : i0d + 1; float fd = pd - (float)i0d;

  #define AV(ih, iw, id) aw[(((ih) * 3 + (iw)) * 3 + (id)) * 8]
  float v00 = AV(i0h, i0w, i0d) * (1.f - fd) + AV(i0h, i0w, i1d) * fd;
  float v01 = AV(i0h, i1w, i0d) * (1.f - fd) + AV(i0h, i1w, i1d) * fd;
  float v10 = AV(i1h, i0w, i0d) * (1.f - fd) + AV(i1h, i0w, i1d) * fd;
  float v11 = AV(i1h, i1w, i0d) * (1.f - fd) + AV(i1h, i1w, i1d) * fd;
  #undef AV
  float v0 = v00 * (1.f - fw) + v01 * fw;
  float v1 = v10 * (1.f - fw) + v11 * fw;
  out[idx] = v0 * (1.f - fh) + v1 * fh;
}

extern "C" void kernel_launch(void* const* d_in, const int* in_sizes, int n_in,
                              void* d_out, int out_size, void* d_ws, size_t ws_size,
                              hipStream_t stream) {
  const float* q   = (const float*)d_in[0];
  const float* k   = (const float*)d_in[1];
  const float* v   = (const float*)d_in[2];
  const float* rpb = (const float*)d_in[3];

  float* ws = (float*)d_ws;
  const size_t TOT = (size_t)TOT_PER_TENSOR;
  float* qg = ws;
  float* kg = ws + TOT;
  float* vg = ws + 2 * TOT;
  float* at = ws + 3 * TOT;

  {
    long long total = 3LL * (long long)TOT;
    int blocks = (int)((total + 255) / 256);
    gather_pool_kernel<<<blocks, 256, 0, stream>>>(q, k, v, qg, kg, vg);
  }
  {
    int blocks = (4 * 2 * NWIN_TOT) / 4;   // 9360 blocks, 4 waves each
    win_attn_wmma_kernel<<<blocks, 128, 0, stream>>>(qg, kg, vg, rpb, at);
  }
  {
    long long total = 4LL * 64 * 110592;
    int blocks = (int)((total + 255) / 256);
    scatter_upsample_kernel<<<blocks, 256, 0, stream>>>(at, (float*)d_out);
  }
}